// Gnn_40845138985251
// MI455X (gfx1250) — compile-verified
//
#include <hip/hip_runtime.h>
#include <math.h>

// ---------------------------------------------------------------------------
// GATv2 x2 on MI455X (gfx1250, wave32).
// Dense node projections use V_WMMA_F32_16X16X4_F32 (fp32-native WMMA).
// Edge passes are L2-resident gather + atomic scatter (GLOBAL_ATOMIC_*).
// ---------------------------------------------------------------------------

typedef __attribute__((ext_vector_type(2))) float v2f;
typedef __attribute__((ext_vector_type(8))) float v8f;

#define NEG_SLOPE 0.2f

// --- ordered-float <-> uint key for exact atomic segment-max -----------------
__device__ __forceinline__ unsigned fkey(float f) {
  unsigned u = __float_as_uint(f);
  return (u & 0x80000000u) ? ~u : (u | 0x80000000u);
}
__device__ __forceinline__ float funkey(unsigned u) {
  unsigned v = (u & 0x80000000u) ? (u & 0x7FFFFFFFu) : ~u;
  return __uint_as_float(v);
}

// ---------------------------------------------------------------------------
// Pass A: per-target-node mean of raw edge_attr (self-loop fill_value='mean')
// ---------------------------------------------------------------------------
__global__ void gat_count_sum(const int* __restrict__ ei,
                              const float* __restrict__ eattr,
                              float* __restrict__ cnt,
                              float* __restrict__ lattr, int E) {
  int e = blockIdx.x * blockDim.x + threadIdx.x;
  if (e >= E) return;
  int dst = ei[E + e];
  const float* ea = eattr + (size_t)e * 4;
  atomicAdd(cnt + dst, 1.0f);
  float* la = lattr + (size_t)dst * 4;
  atomicAdd(la + 0, ea[0]);
  atomicAdd(la + 1, ea[1]);
  atomicAdd(la + 2, ea[2]);
  atomicAdd(la + 3, ea[3]);
}

__global__ void gat_mean(const float* __restrict__ cnt,
                         float* __restrict__ lattr, int N) {
  int i = blockIdx.x * blockDim.x + threadIdx.x;
  if (i >= N) return;
  float inv = 1.0f / fmaxf(cnt[i], 1.0f);
  float* la = lattr + (size_t)i * 4;
  la[0] *= inv; la[1] *= inv; la[2] *= inv; la[3] *= inv;
}

// ---------------------------------------------------------------------------
// Dual node projection via V_WMMA_F32_16X16X4_F32.
//   Ya = X @ Wa,  Yb = X @ Wb      X:[n,DIN] Wa/Wb:[DIN,DOUT] row-major
// One wave computes a 16-row tile. n is a multiple of 16 (100000 = 6250*16),
// so EXEC is all-ones inside every executing wave (WMMA requirement).
// A layout (16x4 f32): lanes 0-15 hold K=0,1; lanes 16-31 hold K=2,3.
// C/D layout (16x16 f32): VGPR r -> row r (lanes 0-15) / row 8+r (lanes 16-31).
// ---------------------------------------------------------------------------
template <int DIN, int DOUT>
__global__ void proj2_wmma(const float* __restrict__ X,
                           const float* __restrict__ Wa,
                           const float* __restrict__ Wb,
                           float* __restrict__ Ya,
                           float* __restrict__ Yb, int n) {
  const int NT = DOUT / 16;
  int wave = (blockIdx.x * blockDim.x + threadIdx.x) >> 5;
  int lane = threadIdx.x & 31;
  int row0 = wave * 16;
  if (row0 >= n) return;            // whole-wave uniform: EXEC stays all-1s
  int half = lane >> 4;             // 0: K pair {0,1}; 1: K pair {2,3}
  int l = lane & 15;

  v8f ca[NT], cb[NT];
#pragma unroll
  for (int nt = 0; nt < NT; ++nt) {
#pragma unroll
    for (int r = 0; r < 8; ++r) { ca[nt][r] = 0.0f; cb[nt][r] = 0.0f; }
  }

#pragma unroll
  for (int k0 = 0; k0 < DIN; k0 += 4) {
    int kr = k0 + 2 * half;
    const float* xp = X + (size_t)(row0 + l) * DIN + kr;
    v2f a; a.x = xp[0]; a.y = xp[1];
#pragma unroll
    for (int nt = 0; nt < NT; ++nt) {
      int col = nt * 16 + l;
      v2f ba, bb;
      ba.x = Wa[(size_t)kr * DOUT + col];
      ba.y = Wa[(size_t)(kr + 1) * DOUT + col];
      bb.x = Wb[(size_t)kr * DOUT + col];
      bb.y = Wb[(size_t)(kr + 1) * DOUT + col];
      ca[nt] = __builtin_amdgcn_wmma_f32_16x16x4_f32(
          false, a, false, ba, (short)0, ca[nt], false, false);
      cb[nt] = __builtin_amdgcn_wmma_f32_16x16x4_f32(
          false, a, false, bb, (short)0, cb[nt], false, false);
    }
  }

#pragma unroll
  for (int nt = 0; nt < NT; ++nt) {
#pragma unroll
    for (int r = 0; r < 8; ++r) {
      int row = row0 + half * 8 + r;
      Ya[(size_t)row * DOUT + nt * 16 + l] = ca[nt][r];
      Yb[(size_t)row * DOUT + nt * 16 + l] = cb[nt][r];
    }
  }
}

// ---------------------------------------------------------------------------
// Edge pass 1: score_e = att . leaky_relu(xl[src] + xr[dst] + We^T ea_e)
// and exact segment-max over dst via ordered-key atomicMax.
// xl/xr gathers are L2-resident (12.8 MB arrays vs 192 MB L2).
// ---------------------------------------------------------------------------
template <int D>
__global__ void gat_score(const int* __restrict__ ei,
                          const float* __restrict__ eattr,
                          const float* __restrict__ lattr,
                          const float* __restrict__ xl,
                          const float* __restrict__ xr,
                          const float* __restrict__ We,   // [4, D]
                          const float* __restrict__ att,  // [D]
                          float* __restrict__ score,
                          unsigned* __restrict__ nmax, int E, int N) {
  int e = blockIdx.x * blockDim.x + threadIdx.x;
  int E2 = E + N;
  if (e >= E2) return;
  int src, dst;
  const float* ea;
  if (e < E) {
    src = ei[e];
    dst = ei[E + e];
    ea = eattr + (size_t)e * 4;
  } else {
    src = dst = e - E;                    // self-loop
    ea = lattr + (size_t)src * 4;
  }
  float e0 = ea[0], e1 = ea[1], e2 = ea[2], e3 = ea[3];
  const float* pl = xl + (size_t)src * D;
  const float* pr = xr + (size_t)dst * D;
  float s = 0.0f;
#pragma unroll
  for (int d = 0; d < D; d += 4) {
    float4 a = *(const float4*)(pl + d);
    float4 b = *(const float4*)(pr + d);
    float va[4] = {a.x, a.y, a.z, a.w};
    float vb[4] = {b.x, b.y, b.z, b.w};
#pragma unroll
    for (int j = 0; j < 4; ++j) {
      float v = va[j] + vb[j] + e0 * We[d + j] + e1 * We[D + d + j] +
                e2 * We[2 * D + d + j] + e3 * We[3 * D + d + j];
      v = v > 0.0f ? v : NEG_SLOPE * v;   // leaky_relu
      s = fmaf(att[d + j], v, s);
    }
  }
  score[e] = s;
  atomicMax(nmax + dst, fkey(s));
}

// ---------------------------------------------------------------------------
// Edge pass 2: w = exp(s - max); denom[dst] += w; acc[dst] += w * xl[src]
// (normalization folded into the final per-node divide).
// ---------------------------------------------------------------------------
template <int D>
__global__ void gat_accum(const int* __restrict__ ei,
                          const float* __restrict__ xl,
                          const float* __restrict__ score,
                          const unsigned* __restrict__ nmax,
                          float* __restrict__ denom,
                          float* __restrict__ acc, int E, int N) {
  int e = blockIdx.x * blockDim.x + threadIdx.x;
  int E2 = E + N;
  if (e >= E2) return;
  int src, dst;
  if (e < E) { src = ei[e]; dst = ei[E + e]; }
  else       { src = dst = e - E; }
  float m = funkey(nmax[dst]);
  float w = __expf(score[e] - m);
  atomicAdd(denom + dst, w);
  const float* pl = xl + (size_t)src * D;
  float* pa = acc + (size_t)dst * D;
#pragma unroll
  for (int d = 0; d < D; d += 4) {
    float4 a = *(const float4*)(pl + d);
    atomicAdd(pa + d + 0, w * a.x);
    atomicAdd(pa + d + 1, w * a.y);
    atomicAdd(pa + d + 2, w * a.z);
    atomicAdd(pa + d + 3, w * a.w);
  }
}

// ---------------------------------------------------------------------------
// Finalize: out = acc/denom + bias (denom > 0 always: every node self-loops)
// ---------------------------------------------------------------------------
template <int D, bool RELU>
__global__ void gat_final(const float* __restrict__ acc,
                          const float* __restrict__ denom,
                          const float* __restrict__ bias,
                          float* __restrict__ out, int N) {
  int i = blockIdx.x * blockDim.x + threadIdx.x;
  if (i >= N * D) return;
  int node = i / D;
  int d = i - node * D;
  float v = acc[i] / denom[node] + bias[d];
  if (RELU) v = fmaxf(v, 0.0f);
  out[i] = v;
}

// ---------------------------------------------------------------------------
extern "C" void kernel_launch(void* const* d_in, const int* in_sizes, int n_in,
                              void* d_out, int out_size, void* d_ws,
                              size_t ws_size, hipStream_t stream) {
  const float* x     = (const float*)d_in[0];
  const int*   ei    = (const int*)d_in[1];
  const float* eattr = (const float*)d_in[2];
  const float* Wl1   = (const float*)d_in[3];
  const float* Wr1   = (const float*)d_in[4];
  const float* We1   = (const float*)d_in[5];
  const float* att1  = (const float*)d_in[6];
  const float* b1    = (const float*)d_in[7];
  const float* Wl2   = (const float*)d_in[8];
  const float* Wr2   = (const float*)d_in[9];
  const float* We2   = (const float*)d_in[10];
  const float* att2  = (const float*)d_in[11];
  const float* b2    = (const float*)d_in[12];

  const int N = in_sizes[0] / 8;   // D_IN = 8
  const int E = in_sizes[2] / 4;   // D_EDGE = 4
  const int E2 = E + N;

  // ---- workspace carve (floats) ----
  float* ws    = (float*)d_ws;
  float* cnt   = ws;                           // N
  float* lattr = cnt + N;                      // 4N
  float* xl    = lattr + (size_t)4 * N;        // 32N
  float* xr    = xl + (size_t)32 * N;          // 32N
  float* h     = xr + (size_t)32 * N;          // 32N
  float* acc   = h + (size_t)32 * N;           // 32N
  float* denom = acc + (size_t)32 * N;         // N
  unsigned* nmax = (unsigned*)(denom + N);     // N  (key 0 == -inf)
  float* score = (float*)(nmax + N);           // E + N

  const int B = 256;
  const int gE  = (E + B - 1) / B;
  const int gE2 = (E2 + B - 1) / B;
  const int gN  = (N + B - 1) / B;
  const int projThreads = ((N + 15) / 16) * 32;     // one wave per 16 rows
  const int gProj = (projThreads + B - 1) / B;

  // ---- self-loop edge_attr = per-dst mean of raw edge_attr ----
  hipMemsetAsync(cnt, 0, sizeof(float) * (size_t)5 * N, stream);
  gat_count_sum<<<gE, B, 0, stream>>>(ei, eattr, cnt, lattr, E);
  gat_mean<<<gN, B, 0, stream>>>(cnt, lattr, N);

  // ================= Layer 1 (8 -> 32, relu) =================
  proj2_wmma<8, 32><<<gProj, B, 0, stream>>>(x, Wl1, Wr1, xl, xr, N);
  hipMemsetAsync(acc, 0, sizeof(float) * (size_t)34 * N, stream); // acc|denom|nmax
  gat_score<32><<<gE2, B, 0, stream>>>(ei, eattr, lattr, xl, xr, We1, att1,
                                       score, nmax, E, N);
  gat_accum<32><<<gE2, B, 0, stream>>>(ei, xl, score, nmax, denom, acc, E, N);
  gat_final<32, true><<<(N * 32 + B - 1) / B, B, 0, stream>>>(acc, denom, b1,
                                                              h, N);

  // ================= Layer 2 (32 -> 16) =================
  proj2_wmma<32, 16><<<gProj, B, 0, stream>>>(h, Wl2, Wr2, xl, xr, N);
  hipMemsetAsync(acc, 0, sizeof(float) * (size_t)34 * N, stream);
  gat_score<16><<<gE2, B, 0, stream>>>(ei, eattr, lattr, xl, xr, We2, att2,
                                       score, nmax, E, N);
  gat_accum<16><<<gE2, B, 0, stream>>>(ei, xl, score, nmax, denom, acc, E, N);
  gat_final<16, false><<<(N * 16 + B - 1) / B, B, 0, stream>>>(
      acc, denom, b2, (float*)d_out, N);
}